// GNNLatencyPredictor_81088982548481
// MI455X (gfx1250) — compile-verified
//
#include <hip/hip_runtime.h>

typedef __attribute__((ext_vector_type(2))) float v2f;
typedef __attribute__((ext_vector_type(4))) float v4f;
typedef __attribute__((ext_vector_type(8))) float v8f;

#define HID 16

// ---------------------------------------------------------------- zero ws
__global__ void k_zero(float* __restrict__ p, int n) {
  int i = blockIdx.x * blockDim.x + threadIdx.x;
  if (i < n) p[i] = 0.0f;
}

// ------------------------------------------------- degree: deg[dst] += 1
__global__ void k_degree(const int* __restrict__ ei, float* __restrict__ deg, int E) {
  int e = blockIdx.x * blockDim.x + threadIdx.x;
  if (e < E) atomicAdd(&deg[ei[E + e]], 1.0f);
}

// ------------------------------------- in place: isq[i] = rsqrt(deg[i]+1)
__global__ void k_invsqrt(float* __restrict__ isq, int n) {
  int i = blockIdx.x * blockDim.x + threadIdx.x;
  if (i < n) isq[i] = rsqrtf(isq[i] + 1.0f);
}

// --------------------------------------------- h1 = x @ W1 via f32 WMMA
// One wave -> 16 nodes x 16 hidden. A = 16x4 (IN_DIM=2 padded), B = 4x16.
// A layout: lanes 0-15 row M=lane, v0=K0, v1=K1; lanes 16-31 K=2,3 (zero pad).
// B layout: lanes 0-15 col N=lane, v0=K0, v1=K1; lanes 16-31 K=2,3 (zero pad).
// D layout: vgpr r -> row r (lanes 0-15) / row 8+r (lanes 16-31), col = lane%16.
__global__ void k_xw1_wmma(const float* __restrict__ x, const float* __restrict__ W1,
                           float* __restrict__ h1, int n) {
  int gid  = blockIdx.x * blockDim.x + threadIdx.x;
  int wave = gid >> 5;
  int lane = threadIdx.x & 31;
  int base = wave * 16;
  if (base >= n) return;               // wave-uniform: EXEC stays all-1s

  int  m    = lane & 15;
  bool lo   = lane < 16;
  bool full = (base + 16 <= n);        // wave-uniform fast path
  float a0 = 0.0f, a1 = 0.0f, b0 = 0.0f, b1 = 0.0f;
  int row = base + m;
  if (lo) {
    b0 = W1[m];                        // W1[0][m]
    b1 = W1[HID + m];                  // W1[1][m]
    if (full || row < n) { a0 = x[row * 2 + 0]; a1 = x[row * 2 + 1]; }
  }
  v2f A = {a0, a1};
  v2f B = {b0, b1};
  v8f C = {};
  C = __builtin_amdgcn_wmma_f32_16x16x4_f32(false, A, false, B, (short)0, C, false, false);

  float* outp = h1 + (size_t)base * HID + m;
  if (full) {
    // straight-line coalesced stores: each r writes two contiguous 64B rows
#pragma unroll
    for (int r = 0; r < 8; ++r) {
      int mr = lo ? r : (8 + r);
      outp[mr * HID] = C[r];
    }
  } else {
#pragma unroll
    for (int r = 0; r < 8; ++r) {
      int mr = lo ? r : (8 + r);
      if (base + mr < n) outp[mr * HID] = C[r];
    }
  }
}

// ---------------- layer-1 aggregation: agg1[dst][d] += c * h1[src][d]
// one thread per (edge, dim): lanes 0..15 read h1[src*16+0..15] coalesced;
// h1/agg1/isq are L2-resident (<= 6.4MB each), atomics resolve in L2.
__global__ void k_agg1(const int* __restrict__ ei, const float* __restrict__ isq,
                       const float* __restrict__ h1, float* __restrict__ agg1, int E) {
  int gid = blockIdx.x * blockDim.x + threadIdx.x;
  int e = gid >> 4;
  int d = gid & 15;
  if (e >= E) return;
  int s = ei[e];
  int t = ei[E + e];
  float c = isq[s] * isq[t];
  atomicAdd(&agg1[t * HID + d], c * h1[s * HID + d]);
}

// ---- finalize layer 1: relu(agg + h*selfc + b1), then dot with W2 -> h2
__global__ void k_layer1_finish(const float* __restrict__ agg1, const float* __restrict__ h1,
                                const float* __restrict__ isq, const float* __restrict__ b1v,
                                const float* __restrict__ W2, float* __restrict__ h2, int n) {
  int i = blockIdx.x * blockDim.x + threadIdx.x;
  if (i >= n) return;
  float s  = isq[i];
  float sc = s * s;
  const v4f* ag = (const v4f*)(agg1 + (size_t)i * HID);
  const v4f* hh = (const v4f*)(h1   + (size_t)i * HID);
  float acc = 0.0f;
#pragma unroll
  for (int q = 0; q < 4; ++q) {
    v4f a = ag[q];
    v4f h = hh[q];
#pragma unroll
    for (int j = 0; j < 4; ++j) {
      int d = q * 4 + j;
      float v = a[j] + h[j] * sc + b1v[d];
      v = fmaxf(v, 0.0f);
      acc = fmaf(v, W2[d], acc);
    }
  }
  h2[i] = acc;
}

// -------------- layer-2 aggregation: agg2[dst] += c * h2[src]
__global__ void k_agg2(const int* __restrict__ ei, const float* __restrict__ isq,
                       const float* __restrict__ h2, float* __restrict__ agg2, int E) {
  int e = blockIdx.x * blockDim.x + threadIdx.x;
  if (e >= E) return;
  int s = ei[e];
  int t = ei[E + e];
  atomicAdd(&agg2[t], isq[s] * isq[t] * h2[s]);
}

// -------------- final: out = agg2 + h2*selfc + b2
__global__ void k_out(const float* __restrict__ agg2, const float* __restrict__ h2,
                      const float* __restrict__ isq, const float* __restrict__ b2,
                      float* __restrict__ out, int n) {
  int i = blockIdx.x * blockDim.x + threadIdx.x;
  if (i >= n) return;
  float s = isq[i];
  out[i] = agg2[i] + h2[i] * s * s + b2[0];
}

extern "C" void kernel_launch(void* const* d_in, const int* in_sizes, int n_in,
                              void* d_out, int out_size, void* d_ws, size_t ws_size,
                              hipStream_t stream) {
  const float* x  = (const float*)d_in[0];
  const int*   ei = (const int*)d_in[1];       // edge_index [2, E] (row 0 = src, row 1 = dst)
  const float* W1 = (const float*)d_in[2];     // [2, 16] row-major
  const float* b1 = (const float*)d_in[3];     // [16]
  const float* W2 = (const float*)d_in[4];     // [16, 1]
  const float* b2 = (const float*)d_in[5];     // [1]
  float* out = (float*)d_out;

  const int N = in_sizes[0] / 2;               // 100000
  const int E = in_sizes[1] / 2;               // 5000000

  // workspace layout (floats): isq[N] | h1[16N] | agg1[16N] | h2[N] | agg2[N]
  float* ws   = (float*)d_ws;
  float* isq  = ws;
  float* h1   = ws + (size_t)N;
  float* agg1 = ws + (size_t)17 * N;
  float* h2   = ws + (size_t)33 * N;
  float* agg2 = ws + (size_t)34 * N;

  const int B = 256;
  // zero only the accumulated regions (h1/h2 are fully overwritten)
  k_zero<<<(N + B - 1) / B, B, 0, stream>>>(isq, N);
  k_zero<<<(16 * N + B - 1) / B, B, 0, stream>>>(agg1, 16 * N);
  k_zero<<<(N + B - 1) / B, B, 0, stream>>>(agg2, N);

  k_degree<<<(E + B - 1) / B, B, 0, stream>>>(ei, isq, E);
  k_invsqrt<<<(N + B - 1) / B, B, 0, stream>>>(isq, N);

  int waves = (N + 15) / 16;                   // one wave per 16 nodes
  int thr   = waves * 32;
  k_xw1_wmma<<<(thr + B - 1) / B, B, 0, stream>>>(x, W1, h1, N);

  long long t1 = (long long)E * 16;            // one thread per (edge, dim)
  k_agg1<<<(int)((t1 + B - 1) / B), B, 0, stream>>>(ei, isq, h1, agg1, E);

  k_layer1_finish<<<(N + B - 1) / B, B, 0, stream>>>(agg1, h1, isq, b1, W2, h2, N);
  k_agg2<<<(E + B - 1) / B, B, 0, stream>>>(ei, isq, h2, agg2, E);
  k_out<<<(N + B - 1) / B, B, 0, stream>>>(agg2, h2, isq, b2, out, N);
}